// CapsLayer_33801392620029
// MI455X (gfx1250) — compile-verified
//
#include <hip/hip_runtime.h>
#include <math.h>

// Problem constants (from reference):
//   x: [B=32, IN_CAPS=2048, IN_DIM=16] f32  -> flat x[b*32768 + i*16 + k]
//   W: [2048, 32, 32, 16] f32              -> flat W[(i*1024 + n)*16 + k], n=j*32+l
//   out vj: [32, 1, 32, 32] f32 = squash over l of s[b,n] = sum_{i,k} W*x
//   (routing loop collapses: softmax over size-1 axis => cj == 1 every round)
#define IN_CAPS  2048
#define IBLK     4           // i-values staged in LDS per step

typedef __attribute__((ext_vector_type(2))) float v2f;
typedef __attribute__((ext_vector_type(4))) float v4f;
typedef __attribute__((ext_vector_type(8))) float v8f;

// V_WMMA_F32_16X16X4_F32: D(16x16 f32) = A(16x4 f32) * B(4x16 f32) + C
// A frag (2 VGPR): lanes 0-15 -> m=lane, {K0,K1}; lanes 16-31 -> m=lane-16, {K2,K3}
// B frag (2 VGPR): lanes 0-15 -> n=lane, rows {K0,K1}; lanes 16-31 -> n=lane-16, {K2,K3}
// C/D (8 VGPR): vgpr v: lanes 0-15 -> (m=v, n=lane); lanes 16-31 -> (m=v+8, n=lane-16)
__device__ __forceinline__ v8f wmma4(v2f a, v2f b, v8f c) {
    return __builtin_amdgcn_wmma_f32_16x16x4_f32(
        /*neg_a=*/false, a, /*neg_b=*/false, b,
        /*c_mod=*/(short)0, c, /*reuse_a=*/false, /*reuse_b=*/false);
}

// ---------------------------------------------------------------------------
// Kernel 1: partial GEMM (M=32, N=1024, K=32768), one wave per (ntile, ksplit).
// All 8 waves of a block share one K-split, so the x slice is staged once per
// block (coalesced b128), W tiles per-wave (contiguous 1KB/i, non-temporal).
// Partials written in raw fragment layout: part[unit*512 + tile*256 + lane*8+v].
// ---------------------------------------------------------------------------
__global__ __launch_bounds__(256)
void caps_gemm_partial(const float* __restrict__ x,
                       const float* __restrict__ W,
                       float* __restrict__ part,
                       int S) {
    // Per-wave W region: IBLK i * 16 n * (16 k + 4 pad) dwords = 1280 dwords.
    __shared__ float wlds[8 * IBLK * 16 * 20];   // 40960 B
    // Block-shared x region: 32 b * (IBLK*16 k + 4 pad) dwords, stride 68.
    __shared__ float xlds[32 * 68];              //  8704 B

    const int wave = threadIdx.x >> 5;
    const int lane = threadIdx.x & 31;
    const unsigned u = blockIdx.x * 8u + wave;   // unit = split*64 + ntile
    const int nt    = u & 63;
    const int split = u >> 6;                    // same for all waves in block
    const int chunk = IN_CAPS / S;
    const int i0    = split * chunk;

    const int nl   = lane & 15;
    const int ksel = (lane >= 16) ? 2 : 0;       // lane's K-pair within a quad

    float* wldsw = wlds + wave * (IBLK * 16 * 20);

    v8f acc0 = {};   // batch rows 0..15
    v8f acc1 = {};   // batch rows 16..31

    for (int ib = 0; ib < chunk; ib += IBLK) {
        __syncthreads();   // previous-stage xlds readers done before overwrite
        // ---- stage x[b][i0+ib .. +IBLK-1][0..15] -> xlds[b*68 + c*4] ----
        // 512 float4 total, 256 threads: 2 coalesced b128 loads each.
#pragma unroll
        for (int t = 0; t < 2; ++t) {
            const int idx = threadIdx.x + 256 * t;   // 0..511
            const int b = idx >> 4, c = idx & 15;    // c = il*4 + kq
            v4f v = *((const v4f*)x + ((size_t)b * 8192 + (size_t)(i0 + ib) * 4 + c));
            *(v4f*)&xlds[b * 68 + c * 4] = v;
        }
        // ---- stage W n-tile, IBLK i-slices (1KB contiguous each), NT hint ----
#pragma unroll
        for (int il = 0; il < IBLK; ++il) {
            const int i = i0 + ib + il;
            const v4f* src = (const v4f*)(W + (size_t)i * 16384 + (size_t)nt * 256);
            v4f w0 = __builtin_nontemporal_load(src + lane);
            v4f w1 = __builtin_nontemporal_load(src + lane + 32);
            const int a0i = lane, a1i = lane + 32;   // float4 index in [0,64)
            *(v4f*)&wldsw[il*320 + (a0i >> 2)*20 + (a0i & 3)*4] = w0;
            *(v4f*)&wldsw[il*320 + (a1i >> 2)*20 + (a1i & 3)*4] = w1;
        }
        __syncthreads();   // xlds visible to all waves
        // ---- consume: 4 K-quads per i, 2 WMMAs per quad ----
#pragma unroll
        for (int il = 0; il < IBLK; ++il) {
#pragma unroll
            for (int kq = 0; kq < 4; ++kq) {
                const int ko = kq * 4 + ksel;
                v2f bf = *(const v2f*)&wldsw[il*320 + nl*20 + ko];
                v2f a0 = *(const v2f*)&xlds[nl*68        + il*16 + ko];
                v2f a1 = *(const v2f*)&xlds[(nl+16)*68   + il*16 + ko];
                acc0 = wmma4(a0, bf, acc0);
                acc1 = wmma4(a1, bf, acc1);
            }
        }
    }

    float* p = part + (size_t)u * 512 + (size_t)lane * 8;
    *(v8f*)(p)       = acc0;
    *(v8f*)(p + 256) = acc1;
}

// ---------------------------------------------------------------------------
// Kernel 2: sum partials over splits, de-swizzle fragments, squash, store.
// One block per output capsule j (32 l-values x 32 batches).
// ---------------------------------------------------------------------------
__global__ __launch_bounds__(256)
void caps_reduce_squash(const float* __restrict__ part,
                        float* __restrict__ out,
                        int S) {
    __shared__ float sv[1024];   // sv[b*32 + l]
    __shared__ float sc[32];     // squash scale per b
    const int j = blockIdx.x;

    for (int e = threadIdx.x; e < 1024; e += 256) {
        const int b = e >> 5, l = e & 31;
        const int n = j * 32 + l;
        const int nt = n >> 4, nl2 = n & 15;
        const int tile = b >> 4, mloc = b & 15;
        const int v = mloc & 7;
        const int lane = ((mloc >> 3) << 4) + nl2;
        const size_t base = (size_t)(nt * 2 + tile) * 256 + lane * 8 + v;
        float s = 0.f;
        for (int sp = 0; sp < S; ++sp)
            s += part[base + (size_t)sp * 32768];
        sv[e] = s;
    }
    __syncthreads();

    if (threadIdx.x < 32) {
        const int b = threadIdx.x;
        float ss = 0.f;
#pragma unroll
        for (int l = 0; l < 32; ++l) {
            const float t = sv[b * 32 + l];
            ss += t * t;
        }
        sc[b] = (ss / (1.f + ss)) / sqrtf(ss + 1e-7f);
    }
    __syncthreads();

    for (int e = threadIdx.x; e < 1024; e += 256) {
        const int b = e >> 5, l = e & 31;
        out[(size_t)b * 1024 + j * 32 + l] = sv[e] * sc[b];
    }
}

// ---------------------------------------------------------------------------
extern "C" void kernel_launch(void* const* d_in, const int* in_sizes, int n_in,
                              void* d_out, int out_size, void* d_ws, size_t ws_size,
                              hipStream_t stream) {
    const float* x = (const float*)d_in[0];   // [32,2048,16]
    const float* W = (const float*)d_in[1];   // [2048,32,32,16]
    float* out  = (float*)d_out;              // [32,1,32,32]
    float* part = (float*)d_ws;

    // One K-split needs 64 tiles * 512 floats = 128 KB of workspace.
    int S = 32;
    while (S > 1 && (size_t)S * 64 * 512 * sizeof(float) > ws_size) S >>= 1;

    caps_gemm_partial<<<dim3(8 * S), dim3(256), 0, stream>>>(x, W, part, S);
    caps_reduce_squash<<<dim3(32), dim3(256), 0, stream>>>(part, out, S);
}